// GNN_89842125897936
// MI455X (gfx1250) — compile-verified
//
#include <hip/hip_runtime.h>
#include <hip/hip_bf16.h>

// ---------------------------------------------------------------------------
// GCN forward for MI455X (gfx1250, wave32).
//  - Dense transforms via V_WMMA_F32_16X16X4_F32 (full fp32 matrix pipe).
//  - Edge scatter via edge-parallel global_atomic_add_f32 (memory bound path,
//    ~1.2GB of HBM traffic dominates; GEMMs are ~1.6 GFLOP total).
//  - Degree / symmetric norm computed once, reused by all 3 layers.
// ---------------------------------------------------------------------------

#define HDIM 64
#define GBATCH 256
#define BN_EPS 1e-5f

typedef __attribute__((ext_vector_type(2))) float v2f;
typedef __attribute__((ext_vector_type(8))) float v8f;

// ---------------- degree / norm ----------------
__global__ void k_deg_init(float* deg, int n) {
    int i = blockIdx.x * blockDim.x + threadIdx.x;
    if (i < n) deg[i] = 1.0f;   // self loop
}
__global__ void k_deg_acc(const int* __restrict__ dst, float* deg, int e) {
    int i = blockIdx.x * blockDim.x + threadIdx.x;
    if (i < e) atomicAdd(&deg[dst[i]], 1.0f);
}
__global__ void k_dinv(float* deg, int n) {
    int i = blockIdx.x * blockDim.x + threadIdx.x;
    if (i < n) deg[i] = rsqrtf(deg[i]);   // deg >= 1 always (self loops)
}
__global__ void k_wnorm(const int* __restrict__ src, const int* __restrict__ dst,
                        const float* __restrict__ dinv, float* __restrict__ wnorm, int e) {
    int i = blockIdx.x * blockDim.x + threadIdx.x;
    if (i < e) wnorm[i] = dinv[src[i]] * dinv[dst[i]];
}

// ---------------- input embedding: z0[N,128] ----------------
__global__ void k_embed(const int* __restrict__ type_ids,
                        const float* __restrict__ c, const float* __restrict__ gm,
                        const float* __restrict__ pos, const float* __restrict__ r,
                        const float* __restrict__ vid,
                        const float* __restrict__ W1, const float* __restrict__ b1,
                        const float* __restrict__ W2, const float* __restrict__ b2,
                        float* __restrict__ z, int n) {
    int i = blockIdx.x * blockDim.x + threadIdx.x;       // over N*128
    if (i >= n * 128) return;
    int node = i >> 7, j = i & 127;
    float v;
    if (j < HDIM) {
        // one_hot(type_ids) @ W1 == row gather of W1
        v = W1[type_ids[node] * HDIM + j] + b1[j];
    } else {
        int jj = j - HDIM;
        v = c[node]   * W2[0 * HDIM + jj]
          + gm[node]  * W2[1 * HDIM + jj]
          + pos[node] * W2[2 * HDIM + jj]
          + r[node]   * W2[3 * HDIM + jj]
          + vid[node] * W2[4 * HDIM + jj] + b2[jj];
    }
    z[(size_t)node * 128 + j] = v;
}

// ---------------- WMMA fp32 GEMM: Y[N,64] = Z[N,K] @ W[K,64] ----------------
// One wave per 16x16 output tile; V_WMMA_F32_16X16X4_F32 over K in steps of 4.
// ISA 7.12.2 fp32 operand layouts:
//   A 16x4: lane L(0..15) holds A[L][k0+0..1] in v0/v1, lane L+16 holds A[L][k0+2..3]
//   B 4x16: lane L(0..15) holds B[k0+0..1][L],  lane L+16 holds B[k0+2..3][L]
//   C/D 16x16: vgpr r -> row m0+r (lanes 0-15) / m0+8+r (lanes 16-31), col n0+(lane&15)
__global__ void k_gemm_wmma(const float* __restrict__ Z, int ldz,
                            const float* __restrict__ W,   // [K x 64] row-major
                            float* __restrict__ Y,
                            int K, int nRows, int numTiles) {
    int wave = blockIdx.x * (blockDim.x >> 5) + (threadIdx.x >> 5);
    if (wave >= numTiles) return;                 // uniform per-wave: EXEC stays all-1s
    int lane = threadIdx.x & 31;
    int half = lane >> 4;                         // 0: k0+0..1, 1: k0+2..3
    int l16  = lane & 15;
    int m0 = (wave >> 2) * 16;
    int n0 = (wave & 3) * 16;

    int row = m0 + l16;
    if (row >= nRows) row = nRows - 1;            // clamp (N is a multiple of 16 here)
    const float* zrow = Z + (size_t)row * ldz;
    const float* wcol = W + n0 + l16;

    v8f acc = {0.f, 0.f, 0.f, 0.f, 0.f, 0.f, 0.f, 0.f};
    for (int k0 = 0; k0 < K; k0 += 4) {
        int kk = k0 + half * 2;
        v2f a, b;
        a.x = zrow[kk];
        a.y = zrow[kk + 1];
        b.x = wcol[(size_t)kk * HDIM];
        b.y = wcol[(size_t)(kk + 1) * HDIM];
        // 8 args: (neg_a, A, neg_b, B, c_mod, C, reuse_a, reuse_b)
        acc = __builtin_amdgcn_wmma_f32_16x16x4_f32(false, a, false, b,
                                                    (short)0, acc, false, false);
    }
    int mbase = m0 + half * 8;
    float* yb = Y + (size_t)mbase * HDIM + n0 + l16;
#pragma unroll
    for (int rI = 0; rI < 8; ++rI)
        if (mbase + rI < nRows) yb[(size_t)rI * HDIM] = acc[rI];
}

// ---------------- GCN aggregation ----------------
// H = bias + self-loop term (init), then edge-parallel atomic scatter.
__global__ void k_self(const float* __restrict__ Y, const float* __restrict__ dinv,
                       const float* __restrict__ bias, float* __restrict__ H, int n) {
    int i = blockIdx.x * blockDim.x + threadIdx.x;   // over N*64
    if (i >= n * HDIM) return;
    int node = i >> 6, ch = i & 63;
    float di = dinv[node];
    H[i] = Y[i] * di * di + bias[ch];
}
__global__ void k_edge(const int* __restrict__ src, const int* __restrict__ dst,
                       const float* __restrict__ wnorm,
                       const float* __restrict__ Y, float* __restrict__ H, int e) {
    int i = blockIdx.x * blockDim.x + threadIdx.x;   // over E*64
    if (i >= e * HDIM) return;
    int ed = i >> 6, ch = i & 63;
    int s = src[ed], d = dst[ed];
    atomicAdd(&H[(size_t)d * HDIM + ch], Y[(size_t)s * HDIM + ch] * wnorm[ed]);
}

// ---------------- BatchNorm ----------------
__global__ void k_zero(float* p, int n) {
    int i = blockIdx.x * blockDim.x + threadIdx.x;
    if (i < n) p[i] = 0.0f;
}
__global__ void k_bnstats(const float* __restrict__ H, float* __restrict__ sums, int n) {
    __shared__ float ssum[HDIM], ssq[HDIM];
    int t = threadIdx.x;
    if (t < HDIM) { ssum[t] = 0.f; ssq[t] = 0.f; }
    __syncthreads();
    int total = n * HDIM;
    for (int i = blockIdx.x * blockDim.x + t; i < total; i += gridDim.x * blockDim.x) {
        float v = H[i];
        int ch = i & 63;
        atomicAdd(&ssum[ch], v);
        atomicAdd(&ssq[ch], v * v);
    }
    __syncthreads();
    if (t < HDIM) {
        atomicAdd(&sums[t], ssum[t]);
        atomicAdd(&sums[HDIM + t], ssq[t]);
    }
}
__global__ void k_bnfinal(const float* __restrict__ sums,
                          const float* __restrict__ gamma, const float* __restrict__ beta,
                          float* __restrict__ ss, int n) {
    int t = threadIdx.x;
    if (t >= HDIM) return;
    float inv_n = 1.0f / (float)n;
    float mean = sums[t] * inv_n;
    float var  = sums[HDIM + t] * inv_n - mean * mean;   // population variance
    float sc = gamma[t] * rsqrtf(var + BN_EPS);
    ss[t] = sc;
    ss[HDIM + t] = beta[t] - mean * sc;
}
__global__ void k_bnrelu(const float* __restrict__ H, const float* __restrict__ ss,
                         float* __restrict__ Z, int n) {
    int i = blockIdx.x * blockDim.x + threadIdx.x;
    if (i >= n * HDIM) return;
    int ch = i & 63;
    Z[i] = fmaxf(H[i] * ss[ch] + ss[HDIM + ch], 0.0f);
}

// ---------------- pooling + output head ----------------
__global__ void k_pool(const float* __restrict__ H, const int* __restrict__ batch,
                       float* __restrict__ pooled, int n) {
    int i = blockIdx.x * blockDim.x + threadIdx.x;   // over N*64
    if (i >= n * HDIM) return;
    int node = i >> 6, ch = i & 63;
    float v = fmaxf(H[i], 0.0f);                     // outer relu
    atomicAdd(&pooled[(size_t)batch[node] * HDIM + ch], v);
}
__global__ void k_counts(const int* __restrict__ batch, float* __restrict__ counts, int n) {
    int i = blockIdx.x * blockDim.x + threadIdx.x;
    if (i < n) atomicAdd(&counts[batch[i]], 1.0f);
}
__global__ void k_out(const float* __restrict__ pooled, const float* __restrict__ counts,
                      const float* __restrict__ Wout, const float* __restrict__ bout,
                      float* __restrict__ out) {
    int i = blockIdx.x * blockDim.x + threadIdx.x;   // over G*4
    if (i >= GBATCH * 4) return;
    int g = i >> 2, o = i & 3;
    float inv = 1.0f / fmaxf(counts[g], 1.0f);
    float acc = bout[o];
    const float* pr = pooled + (size_t)g * HDIM;
#pragma unroll 8
    for (int k = 0; k < HDIM; ++k) acc += pr[k] * inv * Wout[k * 4 + o];
    out[i] = acc;
}

// ---------------------------------------------------------------------------
#define CDIV(a, b) (((a) + (b) - 1) / (b))

extern "C" void kernel_launch(void* const* d_in, const int* in_sizes, int n_in,
                              void* d_out, int out_size, void* d_ws, size_t ws_size,
                              hipStream_t stream) {
    const int*   type_ids = (const int*)d_in[0];
    const float* cf   = (const float*)d_in[1];
    const float* gm   = (const float*)d_in[2];
    const float* pos  = (const float*)d_in[3];
    const float* rr   = (const float*)d_in[4];
    const float* vid  = (const float*)d_in[5];
    const int*   ei   = (const int*)d_in[6];
    const int*   batch= (const int*)d_in[7];
    const float* W1   = (const float*)d_in[8];
    const float* b1   = (const float*)d_in[9];
    const float* W2   = (const float*)d_in[10];
    const float* b2   = (const float*)d_in[11];
    const float* Wg0  = (const float*)d_in[12];
    const float* bg0  = (const float*)d_in[13];
    const float* Wg1  = (const float*)d_in[14];
    const float* bg1  = (const float*)d_in[15];
    const float* Wg2  = (const float*)d_in[16];
    const float* bg2  = (const float*)d_in[17];
    const float* gam0 = (const float*)d_in[18];
    const float* bet0 = (const float*)d_in[19];
    const float* gam1 = (const float*)d_in[20];
    const float* bet1 = (const float*)d_in[21];
    const float* Wout = (const float*)d_in[22];
    const float* bout = (const float*)d_in[23];
    float* out = (float*)d_out;

    const int NN = in_sizes[0];           // 50000
    const int EE = in_sizes[6] / 2;       // 800000
    const int* src = ei;
    const int* dst = ei + EE;

    // workspace layout (fp32)
    float* f      = (float*)d_ws;
    float* dinv   = f;                f += NN;
    float* wnorm  = f;                f += EE;
    float* zA     = f;                f += (size_t)NN * 128;   // z0 [N,128], later z1/z2 [N,64]
    float* yB     = f;                f += (size_t)NN * HDIM;  // GEMM output
    float* hC     = f;                f += (size_t)NN * HDIM;  // aggregation output
    float* sums   = f;                f += 2 * HDIM;           // bn raw sums
    float* ss     = f;                f += 2 * HDIM;           // bn scale/shift
    float* pooled = f;                f += GBATCH * HDIM;
    float* counts = f;                f += GBATCH;

    const int B = 256;
    const int numTiles = CDIV(NN, 16) * (HDIM / 16);     // 12500 16x16 tiles
    const int gemmBlocks = CDIV(numTiles, B / 32);       // 8 waves / block

    // --- degrees & symmetric norm (shared by all 3 layers) ---
    k_deg_init<<<CDIV(NN, B), B, 0, stream>>>(dinv, NN);
    k_deg_acc <<<CDIV(EE, B), B, 0, stream>>>(dst, dinv, EE);
    k_dinv    <<<CDIV(NN, B), B, 0, stream>>>(dinv, NN);
    k_wnorm   <<<CDIV(EE, B), B, 0, stream>>>(src, dst, dinv, wnorm, EE);

    // --- embedding ---
    k_embed<<<CDIV(NN * 128, B), B, 0, stream>>>(type_ids, cf, gm, pos, rr, vid,
                                                 W1, b1, W2, b2, zA, NN);

    // --- layer 0: conv(z0[N,128] @ Wg0) -> bn -> relu ---
    k_gemm_wmma<<<gemmBlocks, B, 0, stream>>>(zA, 128, Wg0, yB, 128, NN, numTiles);
    k_self<<<CDIV(NN * HDIM, B), B, 0, stream>>>(yB, dinv, bg0, hC, NN);
    k_edge<<<CDIV(EE * HDIM, B), B, 0, stream>>>(src, dst, wnorm, yB, hC, EE);
    k_zero<<<1, 2 * HDIM, 0, stream>>>(sums, 2 * HDIM);
    k_bnstats<<<512, B, 0, stream>>>(hC, sums, NN);
    k_bnfinal<<<1, HDIM, 0, stream>>>(sums, gam0, bet0, ss, NN);
    k_bnrelu<<<CDIV(NN * HDIM, B), B, 0, stream>>>(hC, ss, zA, NN);

    // --- layer 1 ---
    k_gemm_wmma<<<gemmBlocks, B, 0, stream>>>(zA, HDIM, Wg1, yB, HDIM, NN, numTiles);
    k_self<<<CDIV(NN * HDIM, B), B, 0, stream>>>(yB, dinv, bg1, hC, NN);
    k_edge<<<CDIV(EE * HDIM, B), B, 0, stream>>>(src, dst, wnorm, yB, hC, EE);
    k_zero<<<1, 2 * HDIM, 0, stream>>>(sums, 2 * HDIM);
    k_bnstats<<<512, B, 0, stream>>>(hC, sums, NN);
    k_bnfinal<<<1, HDIM, 0, stream>>>(sums, gam1, bet1, ss, NN);
    k_bnrelu<<<CDIV(NN * HDIM, B), B, 0, stream>>>(hC, ss, zA, NN);

    // --- layer 2 (bare conv) ---
    k_gemm_wmma<<<gemmBlocks, B, 0, stream>>>(zA, HDIM, Wg2, yB, HDIM, NN, numTiles);
    k_self<<<CDIV(NN * HDIM, B), B, 0, stream>>>(yB, dinv, bg2, hC, NN);
    k_edge<<<CDIV(EE * HDIM, B), B, 0, stream>>>(src, dst, wnorm, yB, hC, EE);

    // --- relu + mean pool + head ---
    k_zero<<<CDIV(GBATCH * HDIM + GBATCH, B), B, 0, stream>>>(pooled, GBATCH * HDIM + GBATCH);
    k_pool<<<CDIV(NN * HDIM, B), B, 0, stream>>>(hC, batch, pooled, NN);
    k_counts<<<CDIV(NN, B), B, 0, stream>>>(batch, counts, NN);
    k_out<<<CDIV(GBATCH * 4, B), B, 0, stream>>>(pooled, counts, Wout, bout, out);
}